// Dynamic_MultiTeacher8_52699248721956
// MI455X (gfx1250) — compile-verified
//
#include <hip/hip_runtime.h>
#include <hip/hip_bf16.h>
#include <stdint.h>

// ---------------------------------------------------------------------------
// Multi-teacher KD loss, MI455X (gfx1250).
// - Stage 9 logit rows/sample global->LDS via GLOBAL_LOAD_ASYNC_TO_LDS_B128
//   (ASYNCcnt), compute all row reductions from LDS.
// - mimic = mean of 8 teachers computed with V_WMMA_F32_16X16X4_F32:
//   D(16x16) = A(16x4, all 1/8) x B(4x16 = 4 teacher rows) accumulated twice
//   -> fp32 column sums of the 8xC teacher tile.
// - Tiny deterministic single-block finalize (no float atomics).
// ---------------------------------------------------------------------------

constexpr int NB   = 4096;   // batch
constexpr int NC   = 1000;   // classes
constexpr int CPAD = 1024;   // padded row stride in LDS
constexpr int NV4  = (NC * 4) / 16;   // 250 float4 chunks per row

#define NEG_INF (-__builtin_inff())

typedef float v2f __attribute__((ext_vector_type(2)));
typedef float v8f __attribute__((ext_vector_type(8)));

struct SharedMem {
  float rows[10][CPAD];   // [0..7]=teachers, [8]=student, [9]=mimic
  float red[16];          // cross-wave scratch (per-teacher row max)
};

__device__ __forceinline__ float wave_sum(float v) {
#pragma unroll
  for (int o = 16; o; o >>= 1) v += __shfl_xor(v, o, 32);
  return v;
}
__device__ __forceinline__ float wave_max(float v) {
#pragma unroll
  for (int o = 16; o; o >>= 1) v = fmaxf(v, __shfl_xor(v, o, 32));
  return v;
}

__device__ __forceinline__ void async_row_to_lds(const float* gsrc, const float* lrow, int tid) {
  // one b128 per lane, lanes 0..249 cover 1000 floats
  if (tid < NV4) {
    uint32_t loff = (uint32_t)(size_t)lrow + (uint32_t)tid * 16u;
    unsigned long long ga = (unsigned long long)(size_t)(gsrc + tid * 4);
    asm volatile("global_load_async_to_lds_b128 %0, %1, off"
                 :: "v"(loff), "v"(ga) : "memory");
  }
}

// One (teacher, sample) task done by one wave: top1/top2/target logit and
// the two exp-sums for the KD dot. j = index into all9 order (8 == mimic).
__device__ __forceinline__ void teacher_task(const float* __restrict__ row,
                                             const float* __restrict__ srow,
                                             int tgt, int lane, int b, int j,
                                             float* __restrict__ w_margin,
                                             float* __restrict__ w_tgt,
                                             float* __restrict__ w_q,
                                             float* red8) {
  float m1 = NEG_INF, m2 = NEG_INF, vt = NEG_INF;
  for (int c = lane; c < NC; c += 32) {
    float x = row[c];
    if (x > m1) { m2 = m1; m1 = x; }
    else if (x > m2) { m2 = x; }
    if (c == tgt) vt = x;
  }
#pragma unroll
  for (int o = 16; o; o >>= 1) {
    float o1 = __shfl_xor(m1, o, 32);
    float o2 = __shfl_xor(m2, o, 32);
    float ot = __shfl_xor(vt, o, 32);
    float hi = fmaxf(m1, o1);
    float lo = fminf(m1, o1);
    m2 = fmaxf(lo, fmaxf(m2, o2));
    m1 = hi;
    vt = fmaxf(vt, ot);
  }
  float S = 0.f, P = 0.f;
  for (int c = lane; c < NC; c += 32) {
    float e = __expf((row[c] - m1) * 0.05f);   // 1/T_KD = 0.05
    S += e;
    P += e * srow[c];
  }
  S = wave_sum(S);
  P = wave_sum(P);
  if (lane == 0) {
    w_margin[j * NB + b] = (vt == m1) ? (m1 - m2) : 0.f;
    w_tgt[j * NB + b]    = vt;
    w_q[j * NB + b]      = P / S;      // Sum_c p_tT * s
    if (red8) red8[j] = m1;            // row max, teachers 0..7 only
  }
}

__global__ __launch_bounds__(256)
void kd_pass1(const float* __restrict__ t0, const float* __restrict__ t1,
              const float* __restrict__ t2, const float* __restrict__ t3,
              const float* __restrict__ t4, const float* __restrict__ t5,
              const float* __restrict__ t6, const float* __restrict__ t7,
              const float* __restrict__ s,  const long long* __restrict__ targets,
              float* __restrict__ wsf) {
  __shared__ SharedMem smem;
  const int tid  = threadIdx.x;
  const int lane = tid & 31;
  const int wv   = tid >> 5;
  const int b    = blockIdx.x;

  float* w_margin = wsf;
  float* w_tgt    = wsf + 9 * NB;
  float* w_q      = wsf + 18 * NB;
  float* w_ce     = wsf + 27 * NB;
  float* w_lse    = wsf + 28 * NB;
  float* w_rmax   = wsf + 29 * NB;

  const size_t rowoff = (size_t)b * NC;
  const float* srcs[9] = {t0 + rowoff, t1 + rowoff, t2 + rowoff, t3 + rowoff,
                          t4 + rowoff, t5 + rowoff, t6 + rowoff, t7 + rowoff,
                          s + rowoff};

  // ---- async stage: 9 rows -> LDS ----
#pragma unroll
  for (int a = 0; a < 9; ++a)
    async_row_to_lds(srcs[a], &smem.rows[a][0], tid);
#if __has_builtin(__builtin_amdgcn_s_wait_asynccnt)
  __builtin_amdgcn_s_wait_asynccnt(0);
#else
  asm volatile("s_wait_asynccnt 0x0" ::: "memory");
#endif
  __syncthreads();

  // ---- mimic = mean of 8 teachers ----
#if __has_builtin(__builtin_amdgcn_wmma_f32_16x16x4_f32)
  {
    // Column sums of the 8x16 teacher tile per step:
    // D = A(16x4, all 1/8) x B(4x16 = teachers {0..3}) then += teachers {4..7}.
    // B layout (4x16 f32): vgpr0: lanes0-15 K=0 / lanes16-31 K=2;
    //                      vgpr1: lanes0-15 K=1 / lanes16-31 K=3; N = lane%16.
    const int half = lane >> 4;        // 0 -> K=0,1 ; 1 -> K=2,3
    const int n    = lane & 15;
    v2f a; a.x = 0.125f; a.y = 0.125f;
#pragma unroll
    for (int it = 0; it < 8; ++it) {
      const int base = (it * 8 + wv) * 16;     // 8 waves x 8 tiles x 16 = 1024
      v2f b0, b1;
      b0.x = smem.rows[half * 2 + 0][base + n];
      b0.y = smem.rows[half * 2 + 1][base + n];
      b1.x = smem.rows[half * 2 + 4][base + n];
      b1.y = smem.rows[half * 2 + 5][base + n];
      v8f c = {};
      c = __builtin_amdgcn_wmma_f32_16x16x4_f32(false, a, false, b0, (short)0, c, false, false);
      c = __builtin_amdgcn_wmma_f32_16x16x4_f32(false, a, false, b1, (short)0, c, false, false);
      // D row M=0 lives in D-vgpr0, lanes 0..15 (N = lane)
      if (lane < 16) smem.rows[9][base + lane] = c[0];
    }
  }
#else
  for (int c = tid; c < NC; c += 256) {
    float acc = 0.f;
#pragma unroll
    for (int a = 0; a < 8; ++a) acc += smem.rows[a][c];
    smem.rows[9][c] = acc * 0.125f;
  }
#endif
  __syncthreads();

  const int tgt = (int)targets[b];
  const float* srow = &smem.rows[8][0];

  // ---- waves 0..7 each own teacher j = wv ----
  teacher_task(&smem.rows[wv][0], srow, tgt, lane, b, wv,
               w_margin, w_tgt, w_q, smem.red);
  __syncthreads();

  if (wv == 0) {
    // mimic teacher (all9 index 8)
    teacher_task(&smem.rows[9][0], srow, tgt, lane, b, 8,
                 w_margin, w_tgt, w_q, nullptr);
  } else if (wv == 1) {
    // student stats: CE and logsumexp(s/T_KD)
    float m1 = NEG_INF, vt = NEG_INF;
    for (int c = lane; c < NC; c += 32) {
      float x = srow[c];
      m1 = fmaxf(m1, x);
      if (c == tgt) vt = x;
    }
    m1 = wave_max(m1);
    vt = wave_max(vt);
    float S1 = 0.f, ST = 0.f;
    for (int c = lane; c < NC; c += 32) {
      float x = srow[c];
      S1 += __expf(x - m1);
      ST += __expf((x - m1) * 0.05f);
    }
    S1 = wave_sum(S1);
    ST = wave_sum(ST);
    if (lane == 0) {
      w_ce[b]  = (m1 + logf(S1)) - vt;          // -log_softmax(s)[tgt]
      w_lse[b] = m1 * 0.05f + logf(ST);         // logsumexp(s / 20)
    }
  } else if (wv == 2 && lane == 0) {
    float mm = NEG_INF;
#pragma unroll
    for (int a = 0; a < 8; ++a) mm = fmaxf(mm, smem.red[a]);
    w_rmax[b] = mm;                              // per-sample max over 8 teachers
  }
}

// Single-block deterministic finalize: global max_preds, per-sample 9-way
// threshold softmax + loss combine, fixed-order reduction to the scalar.
__global__ __launch_bounds__(1024)
void kd_finalize(const float* __restrict__ wsf, float* __restrict__ out) {
  const int tid  = threadIdx.x;
  const int lane = tid & 31;
  const int wv   = tid >> 5;
  __shared__ float sred[32];

  const float* w_margin = wsf;
  const float* w_tgt    = wsf + 9 * NB;
  const float* w_q      = wsf + 18 * NB;
  const float* w_ce     = wsf + 27 * NB;
  const float* w_lse    = wsf + 28 * NB;
  const float* w_rmax   = wsf + 29 * NB;

  // ---- global max over 8 teachers ----
  float m = NEG_INF;
  for (int i = tid; i < NB; i += 1024) m = fmaxf(m, w_rmax[i]);
  m = wave_max(m);
  if (lane == 0) sred[wv] = m;
  __syncthreads();
  if (tid == 0) {
    float mm = NEG_INF;
#pragma unroll
    for (int i = 0; i < 32; ++i) mm = fmaxf(mm, sred[i]);
    sred[0] = mm;
  }
  __syncthreads();
  const float inv_maxp = 1.f / sred[0];
  __syncthreads();   // sred reused below

  // ---- per-sample combine ----
  float acc = 0.f;
  for (int b = tid; b < NB; b += 1024) {
    float mg[9], tg[9], qq[9];
#pragma unroll
    for (int j = 0; j < 9; ++j) {
      mg[j] = w_margin[j * NB + b];
      tg[j] = w_tgt[j * NB + b];
      qq[j] = w_q[j * NB + b];
    }
    const float ce  = w_ce[b];
    const float lse = w_lse[b];

    float mm = NEG_INF;
#pragma unroll
    for (int j = 0; j < 9; ++j) mm = fmaxf(mm, mg[j] * 0.5f);   // margin / T_THR
    float es = 0.f, e[9];
#pragma unroll
    for (int j = 0; j < 9; ++j) { e[j] = __expf(mg[j] * 0.5f - mm); es += e[j]; }
    const float inv_es = 1.f / es;

    float lacc = 0.f;
#pragma unroll
    for (int j = 0; j < 9; ++j) {
      float wgt  = tg[j] * inv_maxp;
      float kd   = 400.f * lse - 20.f * qq[j];   // T^2*lse - T*Q
      float loss = (1.f - 0.8f * wgt) * ce + 0.8f * wgt * kd;
      lacc += (e[j] * inv_es) * loss;
    }
    acc += lacc;
  }
  acc *= (1.0f / (float)NB);

  acc = wave_sum(acc);
  if (lane == 0) sred[wv] = acc;
  __syncthreads();
  if (tid == 0) {
    float t = 0.f;
#pragma unroll
    for (int i = 0; i < 32; ++i) t += sred[i];
    out[0] = t;
  }
}

extern "C" void kernel_launch(void* const* d_in, const int* in_sizes, int n_in,
                              void* d_out, int out_size, void* d_ws, size_t ws_size,
                              hipStream_t stream) {
  (void)in_sizes; (void)n_in; (void)out_size; (void)ws_size;
  const float* t0 = (const float*)d_in[0];
  const float* t1 = (const float*)d_in[1];
  const float* t2 = (const float*)d_in[2];
  const float* t3 = (const float*)d_in[3];
  const float* t4 = (const float*)d_in[4];
  const float* t5 = (const float*)d_in[5];
  const float* t6 = (const float*)d_in[6];
  const float* t7 = (const float*)d_in[7];
  const float* s  = (const float*)d_in[8];
  const long long* tgt = (const long long*)d_in[9];
  float* wsf = (float*)d_ws;

  kd_pass1<<<NB, 256, 0, stream>>>(t0, t1, t2, t3, t4, t5, t6, t7, s, tgt, wsf);
  kd_finalize<<<1, 1024, 0, stream>>>(wsf, (float*)d_out);
}